// PrototypeLayer_27504970564139
// MI455X (gfx1250) — compile-verified
//
#include <hip/hip_runtime.h>

typedef __attribute__((ext_vector_type(2))) float v2f;
typedef __attribute__((ext_vector_type(8))) float v8f;

constexpr int N = 32768;
constexpr int P = 2048;
constexpr int D = 256;

// ---------------------------------------------------------------------------
// Kernel 1: squared norms of x rows and prototype rows; init min buffers.
// ---------------------------------------------------------------------------
__global__ void prep_kernel(const float* __restrict__ x,
                            const float* __restrict__ prot,
                            float* __restrict__ xsq,
                            float* __restrict__ psq,
                            unsigned* __restrict__ rowmin,
                            unsigned* __restrict__ colmin)
{
    const int tid  = blockIdx.x * blockDim.x + threadIdx.x;
    if (tid < N) rowmin[tid] = 0x7f800000u;   // +inf
    if (tid < P) colmin[tid] = 0x7f800000u;

    const int wave = tid >> 5;
    const int lane = tid & 31;
    if (wave < N) {
        const float* row = x + (size_t)wave * D;
        float s = 0.f;
        #pragma unroll
        for (int i = 0; i < D / 32; ++i) { float v = row[lane + i * 32]; s += v * v; }
        #pragma unroll
        for (int off = 16; off > 0; off >>= 1) s += __shfl_xor(s, off, 32);
        if (lane == 0) xsq[wave] = s;
    } else if (wave < N + P) {
        const int r = wave - N;
        const float* row = prot + (size_t)r * D;
        float s = 0.f;
        #pragma unroll
        for (int i = 0; i < D / 32; ++i) { float v = row[lane + i * 32]; s += v * v; }
        #pragma unroll
        for (int off = 16; off > 0; off >>= 1) s += __shfl_xor(s, off, 32);
        if (lane == 0) psq[r] = s;
    }
}

// ---------------------------------------------------------------------------
// Kernel 2: tiled fp32 WMMA GEMM with async global->LDS double buffering,
// fused distance epilogue + min reductions.
// Block = 256 threads (8 waves); 128x128 output tile; wave grid 4x2, each
// wave owns 2x4 accumulators of 16x16 (v_wmma_f32_16x16x4_f32).
// ---------------------------------------------------------------------------
__global__ __launch_bounds__(256) void dist_kernel(
        const float* __restrict__ x,
        const float* __restrict__ prot,
        const float* __restrict__ xsq,
        const float* __restrict__ psq,
        float* __restrict__ out,
        unsigned* __restrict__ rowmin,
        unsigned* __restrict__ colmin)
{
    // Row stride padded to 36 floats: multiple of 4 (keeps 16B alignment for
    // B128 transfers) and 36*dr mod 64 distinct for dr=0..15 -> conflict-free
    // 16-row column reads for the WMMA fragments.
    __shared__ float    Xs[2][128][36];
    __shared__ float    Ps[2][128][36];
    __shared__ float    xsq_s[128];
    __shared__ float    psq_s[128];
    __shared__ unsigned rmin_s[128];
    __shared__ unsigned cmin_s[128];

    const int      t      = threadIdx.x;
    const unsigned lane   = t & 31u;
    const unsigned waveId = (unsigned)t >> 5;   // 0..7
    const unsigned waveM  = waveId & 3u;        // 0..3 -> 32-row strips
    const unsigned waveN  = waveId >> 2;        // 0..1 -> 64-col strips
    const unsigned l16    = lane & 15u;
    const unsigned h      = lane >> 4;          // lane half (K-pair / row+8 select)

    const int rowBase = blockIdx.y * 128;
    const int colBase = blockIdx.x * 128;

    if (t < 128) {
        xsq_s[t]  = xsq[rowBase + t];
        rmin_s[t] = 0x7f800000u;
    } else {
        psq_s[t - 128]  = psq[colBase + (t - 128)];
        cmin_s[t - 128] = 0x7f800000u;
    }

    // Issue one 128x32 X tile + 128x32 P tile as 8 async B128 copies per wave
    // (global -> LDS directly, no VGPR round trip; tracked by ASYNCcnt).
    auto issue_tiles = [&](int buf, int kc) {
        #pragma unroll
        for (int i = 0; i < 4; ++i) {
            const int idx = t + i * 256;
            const int r   = idx >> 3;
            const int c4  = (idx & 7) * 4;
            const float* gx = x    + (size_t)(rowBase + r) * D + kc + c4;
            const float* gp = prot + (size_t)(colBase + r) * D + kc + c4;
            const unsigned lx = (unsigned)(size_t)&Xs[buf][r][c4];
            const unsigned lp = (unsigned)(size_t)&Ps[buf][r][c4];
            asm volatile("global_load_async_to_lds_b128 %0, %1, off"
                         :: "v"(lx), "v"((unsigned long long)(size_t)gx) : "memory");
            asm volatile("global_load_async_to_lds_b128 %0, %1, off"
                         :: "v"(lp), "v"((unsigned long long)(size_t)gp) : "memory");
        }
    };

    v8f acc[2][4] = {};

    issue_tiles(0, 0);                       // prologue: chunk 0 -> buffer 0

    for (int i = 0; i < D / 32; ++i) {
        const int buf = i & 1;
        if (i < D / 32 - 1) {
            issue_tiles(buf ^ 1, (i + 1) * 32);          // prefetch next chunk
            // async loads complete in order: <=8 outstanding means the current
            // chunk's 8 copies have landed while the prefetch stays in flight.
            asm volatile("s_wait_asynccnt 0x8" ::: "memory");
        } else {
            asm volatile("s_wait_asynccnt 0x0" ::: "memory");
        }
        __syncthreads();                     // current tile visible block-wide

        #pragma unroll
        for (int k = 0; k < 32; k += 4) {
            const int kk = k + 2 * (int)h;   // lane half picks K pair {0,1} vs {2,3}
            v2f a[2], b[4];
            #pragma unroll
            for (int mt = 0; mt < 2; ++mt) {
                const float* p = &Xs[buf][waveM * 32 + mt * 16 + l16][kk];
                v2f av; av.x = p[0]; av.y = p[1];
                a[mt] = av;
            }
            #pragma unroll
            for (int nt = 0; nt < 4; ++nt) {
                const float* p = &Ps[buf][waveN * 64 + nt * 16 + l16][kk];
                v2f bv; bv.x = p[0]; bv.y = p[1];
                b[nt] = bv;
            }
            #pragma unroll
            for (int mt = 0; mt < 2; ++mt) {
                #pragma unroll
                for (int nt = 0; nt < 4; ++nt) {
                    acc[mt][nt] = __builtin_amdgcn_wmma_f32_16x16x4_f32(
                        false, a[mt], false, b[nt],
                        (short)0, acc[mt][nt], false, false);
                }
            }
        }
        __syncthreads();   // compute done before next prefetch reuses this buffer
    }

    // Fused epilogue: dist = max(x^2 + p^2 - 2*dot, 0); NT store + min reductions.
    const float INF = __uint_as_float(0x7f800000u);
    float cmin[4] = {INF, INF, INF, INF};

    #pragma unroll
    for (int mt = 0; mt < 2; ++mt) {
        #pragma unroll
        for (int v = 0; v < 8; ++v) {
            const int    rLocal = waveM * 32 + mt * 16 + v + 8 * (int)h;
            const float  xs     = xsq_s[rLocal];
            const size_t gr     = (size_t)(rowBase + rLocal);
            float rm = INF;
            #pragma unroll
            for (int nt = 0; nt < 4; ++nt) {
                const int cLocal = waveN * 64 + nt * 16 + (int)l16;
                float dd = fmaxf(xs + psq_s[cLocal] - 2.0f * acc[mt][nt][v], 0.0f);
                // Streaming 256 MiB result: non-temporal so it does not evict
                // the x/prototype working set from L2.
                __builtin_nontemporal_store(dd, &out[gr * P + (size_t)(colBase + cLocal)]);
                rm       = fminf(rm, dd);
                cmin[nt] = fminf(cmin[nt], dd);
            }
            rm = fminf(rm, __shfl_xor(rm, 1, 32));
            rm = fminf(rm, __shfl_xor(rm, 2, 32));
            rm = fminf(rm, __shfl_xor(rm, 4, 32));
            rm = fminf(rm, __shfl_xor(rm, 8, 32));
            if (l16 == 0)
                atomicMin(&rmin_s[rLocal], __float_as_uint(rm)); // dist>=0 -> bit order == float order
        }
    }
    #pragma unroll
    for (int nt = 0; nt < 4; ++nt) {
        float cm = fminf(cmin[nt], __shfl_xor(cmin[nt], 16, 32));
        if (h == 0)
            atomicMin(&cmin_s[waveN * 64 + nt * 16 + l16], __float_as_uint(cm));
    }
    __syncthreads();

    if (t < 128) atomicMin(&rowmin[rowBase + t], rmin_s[t]);
    else         atomicMin(&colmin[colBase + (t - 128)], cmin_s[t - 128]);
}

// ---------------------------------------------------------------------------
// Kernel 3: reduce min buffers to r1 (mean of col mins) and r2 (mean of row mins).
// ---------------------------------------------------------------------------
__global__ void finalize_kernel(const unsigned* __restrict__ rowmin,
                                const unsigned* __restrict__ colmin,
                                float* __restrict__ scalars)
{
    __shared__ float sdata[256];
    const int t = threadIdx.x;

    float s = 0.f;
    for (int i = t; i < P; i += 256) s += __uint_as_float(colmin[i]);
    sdata[t] = s; __syncthreads();
    for (int st = 128; st > 0; st >>= 1) {
        if (t < st) sdata[t] += sdata[t + st];
        __syncthreads();
    }
    if (t == 0) scalars[0] = sdata[0] / (float)P;   // r1
    __syncthreads();

    float s2 = 0.f;
    for (int i = t; i < N; i += 256) s2 += __uint_as_float(rowmin[i]);
    sdata[t] = s2; __syncthreads();
    for (int st = 128; st > 0; st >>= 1) {
        if (t < st) sdata[t] += sdata[t + st];
        __syncthreads();
    }
    if (t == 0) scalars[1] = sdata[0] / (float)N;   // r2
}

// ---------------------------------------------------------------------------
extern "C" void kernel_launch(void* const* d_in, const int* in_sizes, int n_in,
                              void* d_out, int out_size, void* d_ws, size_t ws_size,
                              hipStream_t stream)
{
    const float* x    = (const float*)d_in[0];
    const float* prot = (const float*)d_in[1];
    float*       out  = (float*)d_out;

    // Workspace layout (floats/uints): xsq[N] | psq[P] | rowmin[N] | colmin[P]
    float*    xsq    = (float*)d_ws;
    float*    psq    = xsq + N;
    unsigned* rowmin = (unsigned*)(psq + P);
    unsigned* colmin = rowmin + N;

    const int prepThreads = (N + P) * 32;           // one wave per row
    prep_kernel<<<prepThreads / 256, 256, 0, stream>>>(x, prot, xsq, psq, rowmin, colmin);

    dim3 grid(P / 128, N / 128);                    // 16 x 256 blocks
    dist_kernel<<<grid, 256, 0, stream>>>(x, prot, xsq, psq, out, rowmin, colmin);

    finalize_kernel<<<1, 256, 0, stream>>>(rowmin, colmin, out + (size_t)N * P);
}